// CriticalityDistillation_54159537602781
// MI455X (gfx1250) — compile-verified
//
#include <hip/hip_runtime.h>
#include <hip/hip_bf16.h>
#include <cmath>

typedef float f4  __attribute__((ext_vector_type(4)));
typedef float v2f __attribute__((ext_vector_type(2)));
typedef float v8f __attribute__((ext_vector_type(8)));
typedef int   v4i __attribute__((vector_size(16)));   // matches builtin proto

#define DECAY     0.99f
#define HALF_LIFE 256.0f

// ---------------- CDNA5 async global->LDS helpers ----------------------------
// GLOBAL_LOAD_ASYNC_TO_LDS_B128 copies 16B per lane from a per-lane global
// address to a per-lane LDS address, tracked with ASYNCcnt (CDNA5 ISA).
// Probe result (round 2 diagnostic): param0 is 'v4i addrspace(1)*' (global
// src, non-const); so the call is (global_src, lds_dst, imm_offset, imm_cpol).
#if defined(__HIP_DEVICE_COMPILE__) && \
    __has_builtin(__builtin_amdgcn_global_load_async_to_lds_b128)
#define ASYNC_LDS 1
#endif

typedef __attribute__((address_space(1))) v4i* g_v4i_p;   // prints as "__device__"
typedef __attribute__((address_space(3))) v4i* l_v4i_p;   // LDS, 32-bit offset

static __device__ __forceinline__ void compiler_fence() {
    asm volatile("" ::: "memory");
}

#if defined(ASYNC_LDS)
static __device__ __forceinline__ void async_copy16(const float* g, float* l) {
    // AS1 bits == generic bits for global; AS3 offset == low 32 bits of a
    // generic LDS pointer (flat aperture).
    g_v4i_p gp = (g_v4i_p)(uintptr_t)g;
    l_v4i_p lp = (l_v4i_p)(uint32_t)(uintptr_t)l;
    __builtin_amdgcn_global_load_async_to_lds_b128(gp, lp, 0, 0);
}
static __device__ __forceinline__ void wait_async(int n) {
#if __has_builtin(__builtin_amdgcn_s_wait_asynccnt)
    if (n == 0) __builtin_amdgcn_s_wait_asynccnt(0);
    else        __builtin_amdgcn_s_wait_asynccnt(1);
#else
    if (n == 0) asm volatile("s_wait_asynccnt 0x0" ::: "memory");
    else        asm volatile("s_wait_asynccnt 0x1" ::: "memory");
#endif
    compiler_fence();
}
#endif

// ---------------- Pass A: exact top-k selection via rank counting ------------
// rank(i) = #{j: p[j] > p[i]} + #{j < i: p[j] == p[i]}  (matches lax.top_k ties)
// selected iff rank < k; slot == rank -> deterministic compaction, no atomics.
__global__ void rank_topk_kernel(const float* __restrict__ pressure,
                                 int total, int k,
                                 int* __restrict__ event_pos) {
    __shared__ float sp[8192];
    const int tid = threadIdx.x;
    for (int j = tid; j < total; j += blockDim.x) sp[j] = pressure[j];
    __syncthreads();
    const int i = blockIdx.x * blockDim.x + tid;
    if (i >= total) return;
    const float p = sp[i];
    int rank = 0;
#pragma unroll 8
    for (int j = 0; j < total; ++j) {
        const float v = sp[j];
        rank += (v > p) ? 1 : 0;
        rank += ((v == p) && (j < i)) ? 1 : 0;
    }
    if (rank < k) event_pos[rank] = i;
}

// ---------------- Pass P1: per-timestep weight c(u) --------------------------
// sum_t fe(t) = sum_u sq(u) * c(u),  c(u) = sum_{t=max(0,u-H)}^{u-1} 1/min(H,T-1-t)
__global__ void cu_weights_kernel(const int* __restrict__ horizon,
                                  int T, float* __restrict__ cw) {
    const int u = blockIdx.x * blockDim.x + threadIdx.x;
    if (u >= T) return;
    const int H = *horizon;
    int t0 = u - H; if (t0 < 0) t0 = 0;
    double c = 0.0;
    for (int t = t0; t < u; ++t) {
        int cnt = T - 1 - t; if (cnt > H) cnt = H;
        c += 1.0 / (double)cnt;
    }
    cw[u] = (float)c;
}

// ---------------- Pass P2: bank weights + per-layer weight sums --------------
__global__ void bank_weights_kernel(const int* __restrict__ bank_step,
                                    const int* __restrict__ curstep,
                                    int TTL,
                                    float* __restrict__ w,
                                    float* __restrict__ wsum) {
    __shared__ float red[256];
    const int l  = blockIdx.x;
    const int cs = *curstep;
    float acc = 0.f;
    for (int s = threadIdx.x; s < TTL; s += blockDim.x) {
        const int bs = bank_step[l * TTL + s];
        const float valid = (bs >= 0) ? 1.f : 0.f;
        int agei = cs - bs; if (agei < 0) agei = 0;
        const float wt = exp2f(-((float)agei) / HALF_LIFE) * valid;
        w[l * TTL + s] = wt;
        acc += wt;
    }
    red[threadIdx.x] = acc;
    __syncthreads();
    for (int o = 128; o > 0; o >>= 1) {
        if ((int)threadIdx.x < o) red[threadIdx.x] += red[threadIdx.x + o];
        __syncthreads();
    }
    if (threadIdx.x == 0) wsum[l] = red[0];
}

// ---------------- Pass B: single streaming pass over states ------------------
// partial[l,b,uc,d] = sum_{u in chunk} c(u) * states[l,b,u,d]^2
// CDNA5 path: double-buffered GLOBAL_LOAD_ASYNC_TO_LDS_B128 into wave-private
// LDS slices (no cross-wave sync; ASYNCcnt is per-wave), consume via ds_load.
__global__ void __launch_bounds__(256)
sweep_kernel(const float* __restrict__ states,
             const float* __restrict__ cw,
             int B, int T, int D, int UC,
             float* __restrict__ partial) {
    const int uc = blockIdx.x, b = blockIdx.y, l = blockIdx.z;
    const int CH = T / UC;
    const int u0 = uc * CH, uend = u0 + CH;
    const int tid = threadIdx.x;
    const float* __restrict__ base =
        states + ((size_t)(l * B + b) * T) * (size_t)D;
    f4 acc = {};
#if defined(ASYNC_LDS)
    __shared__ __align__(16) float sbuf[2][256 * 4];   // 8 KB, per-thread 16B slots
    // Prologue: bring row u0 in flight.
    async_copy16(base + (size_t)u0 * D + tid * 4, &sbuf[0][tid * 4]);
    for (int u = u0; u < uend; ++u) {
        const int cur = (u - u0) & 1;
        const bool more = (u + 1) < uend;
        if (more)   // overwrites buffer consumed at iteration u-1 (already read)
            async_copy16(base + (size_t)(u + 1) * D + tid * 4,
                         &sbuf[cur ^ 1][tid * 4]);
        wait_async(more ? 1 : 0);          // row u landed in LDS
        const f4 v = *(const f4*)&sbuf[cur][tid * 4];
        acc += (cw[u] * v) * v;
    }
#else
    const int dq = D / 4;
    const f4* __restrict__ src = (const f4*)base;
#pragma unroll 4
    for (int u = u0; u < uend; ++u) {
        const float cu = cw[u];
        const f4 v = __builtin_nontemporal_load(&src[(size_t)u * dq + tid]);
        acc += (cu * v) * v;
    }
#endif
    f4* dst = (f4*)(partial + ((size_t)((l * B + b) * UC + uc)) * (size_t)D);
    dst[tid] = acc;
}

// ---------------- Pass C: exact window mean at the k event positions ---------
__global__ void __launch_bounds__(256)
event_fe_kernel(const float* __restrict__ states,
                const int* __restrict__ event_pos,
                const int* __restrict__ horizon,
                int B, int T, int D, int k,
                float* __restrict__ fe_event) {
    const int j = blockIdx.x, l = blockIdx.y;
    const int H = *horizon;
    const int i = event_pos[j];
    const int b = i / T, t = i - b * T;
    int cnt = T - 1 - t; if (cnt > H) cnt = H;
    const int tid = threadIdx.x;
    const int dq  = D / 4;
    const f4* __restrict__ src =
        (const f4*)(states + ((size_t)(l * B + b) * T) * (size_t)D);
    f4 acc = {};
#pragma unroll 4
    for (int u = t + 1; u <= t + cnt; ++u) {
        // Unconditional speculative row prefetch (global_prefetch_b8):
        // OOB translation failures are silently dropped, so no range guard.
        __builtin_prefetch(&src[(size_t)(u + 8) * dq + tid], 0, 3);
        const f4 v = __builtin_nontemporal_load(&src[(size_t)u * dq + tid]);
        acc += v * v;
    }
    f4 zero = {};
    f4 fe = (cnt > 0) ? (acc / (float)cnt) : zero;
    f4* dst = (f4*)(fe_event + ((size_t)(l * k + j)) * (size_t)D);
    dst[tid] = fe;
}

// ---------------- Pass D: reduce partials, EMA baseline, event evidence ------
__global__ void finalize_kernel(const float* __restrict__ partial,
                                const float* __restrict__ fe_event,
                                const float* __restrict__ baseline,
                                int B, int UC, int D, int k, int total, int L,
                                float* __restrict__ out) {
    const int l = blockIdx.y;
    const int d = blockIdx.x * blockDim.x + threadIdx.x;

    float sa = 0.f;  // sum of fe over ALL (b,t)
    const float* pp = partial + (size_t)l * B * UC * D + d;
    for (int m = 0; m < B * UC; ++m) sa += pp[(size_t)m * D];

    const float* fp = fe_event + (size_t)l * k * D + d;
    float se = 0.f;  // sum of fe over event positions
    for (int j = 0; j < k; ++j) se += fp[(size_t)j * D];

    const float obs = (sa - se) / (float)(total - k);
    const float nb  = DECAY * baseline[(size_t)l * D + d] + (1.f - DECAY) * obs;
    out[(size_t)(L + l) * D + d] = nb;                       // row 1: new_baseline

    float ev = 0.f;
    for (int j = 0; j < k; ++j) ev += fmaxf(fp[(size_t)j * D] - nb, 0.f);
    out[(size_t)l * D + d] = ev / (float)k;                  // row 0: evidence
}

// ---------------- Pass E: score GEMV via V_WMMA_F32_16X16X4_F32 --------------
// For layer l: weighted_ev[d] = sum_s w[l,s] * bank_ev[l,s,d].
// A(16x4): weight row replicated over all 16 M-rows -> all 16 rows of D equal.
// A layout: VGPR0 = K0 (lanes 0-15) / K2 (lanes 16-31); VGPR1 = K1 / K3.
// B layout (4x16): VGPR0 = row K0 / row K2 by lane half; VGPR1 = K1 / K3.
// D row M=0 lives in C[0] on lanes 0-15 (N = lane).
__global__ void __launch_bounds__(256)
score_wmma_kernel(const float* __restrict__ bank_ev,
                  const float* __restrict__ w,
                  const float* __restrict__ wsum,
                  int TTL, int D, int L,
                  float* __restrict__ out) {
    __shared__ float wsh[1024];
    const int l = blockIdx.y;
    for (int s = threadIdx.x; s < TTL; s += blockDim.x) wsh[s] = w[l * TTL + s];
    __syncthreads();

    const int wave = threadIdx.x >> 5;
    const int lane = threadIdx.x & 31;
    const int d0   = blockIdx.x * ((int)blockDim.x / 32) * 16 + wave * 16;
    const int half = lane >> 4;
    const int n    = lane & 15;
    const float* __restrict__ bp = bank_ev + (size_t)l * TTL * (size_t)D;

    v8f c = {};
    for (int s0 = 0; s0 < TTL; s0 += 4) {
        // Unconditional speculative stream-ahead; keeps the inner loop
        // branch-free so the b32/ds loads + v_wmma form a tight clause.
        __builtin_prefetch(&bp[(size_t)(s0 + 16 + 2 * half) * D + d0 + n], 0, 3);
        v2f a, bm;
        a[0]  = wsh[s0 + 2 * half];
        a[1]  = wsh[s0 + 1 + 2 * half];
        bm[0] = bp[(size_t)(s0 + 2 * half) * D + d0 + n];
        bm[1] = bp[(size_t)(s0 + 1 + 2 * half) * D + d0 + n];
#if defined(__HIP_DEVICE_COMPILE__)
        c = __builtin_amdgcn_wmma_f32_16x16x4_f32(false, a, false, bm,
                                                  (short)0, c, false, false);
#else
        c[0] += a[0] * bm[0] + a[1] * bm[1];  // host-pass stub, never executed
#endif
    }
    if (lane < 16) {
        const float ws_ = wsum[l];
        const float sc  = (ws_ > 0.f) ? (c[0] / fmaxf(ws_, 1e-12f)) : 0.f;
        out[(size_t)(2 * L + l) * D + d0 + lane] = sc;       // row 2: score
    }
}

// ---------------- Launcher ---------------------------------------------------
extern "C" void kernel_launch(void* const* d_in, const int* in_sizes, int n_in,
                              void* d_out, int out_size, void* d_ws, size_t ws_size,
                              hipStream_t stream) {
    const float* pressure  = (const float*)d_in[0];
    const float* states    = (const float*)d_in[1];
    const float* bank_ev   = (const float*)d_in[2];
    const float* baseline  = (const float*)d_in[3];
    const int*   bank_step = (const int*)d_in[4];
    const int*   curstep   = (const int*)d_in[5];
    const int*   horizon   = (const int*)d_in[6];
    float* out = (float*)d_out;

    const int Bb    = 4;                       // batch (B) per reference setup
    const int total = in_sizes[0];             // B*T = 8192
    const int T     = total / Bb;              // 2048
    const int LTTL  = in_sizes[4];             // L*TTL
    const int D     = in_sizes[2] / LTTL;      // 1024
    const int L     = in_sizes[3] / D;         // 8
    const int TTL   = LTTL / L;                // 1024
    const int k     = (int)(0.05 * (double)total + 0.5);  // round(EVENT_FRAC*total)=410
    const int UC    = 32;                      // t-chunks for the streaming pass

    // Workspace carve-out (poisoned by harness; every region fully overwritten
    // before it is read, so no memset needed).
    char* ws = (char*)d_ws;
    size_t off = 0;
    auto take = [&](size_t bytes) -> void* {
        void* p = ws + off;
        off += (bytes + 255) & ~(size_t)255;
        return p;
    };
    int*   event_pos = (int*)  take((size_t)k * 4);
    float* cw        = (float*)take((size_t)T * 4);
    float* wbank     = (float*)take((size_t)L * TTL * 4);
    float* wsum      = (float*)take((size_t)L * 4);
    float* partial   = (float*)take((size_t)L * Bb * UC * D * 4);   // 4 MB
    float* fe_event  = (float*)take((size_t)L * k * D * 4);         // 13.4 MB
    (void)ws_size; (void)n_in; (void)out_size;

    rank_topk_kernel  <<<dim3((total + 255) / 256), dim3(256), 0, stream>>>(
        pressure, total, k, event_pos);
    cu_weights_kernel <<<dim3((T + 255) / 256),     dim3(256), 0, stream>>>(
        horizon, T, cw);
    bank_weights_kernel<<<dim3(L),                  dim3(256), 0, stream>>>(
        bank_step, curstep, TTL, wbank, wsum);
    sweep_kernel      <<<dim3(UC, Bb, L),           dim3(D / 4), 0, stream>>>(
        states, cw, Bb, T, D, UC, partial);
    event_fe_kernel   <<<dim3(k, L),                dim3(D / 4), 0, stream>>>(
        states, event_pos, horizon, Bb, T, D, k, fe_event);
    finalize_kernel   <<<dim3(D / 256, L),          dim3(256), 0, stream>>>(
        partial, fe_event, baseline, Bb, UC, D, k, total, L, out);
    score_wmma_kernel <<<dim3(D / 128, L),          dim3(256), 0, stream>>>(
        bank_ev, wbank, wsum, TTL, D, L, out);
}